// GCNNet_NC_37649683316745
// MI455X (gfx1250) — compile-verified
//
#include <hip/hip_runtime.h>
#include <hip/hip_bf16.h>
#include <math.h>

typedef __attribute__((ext_vector_type(2))) float v2f;
typedef __attribute__((ext_vector_type(8))) float v8f;

#define DFEAT 128

// ---------------- degree / normalization ----------------
__global__ void k_init_deg(float* deg, int n) {
  int i = blockIdx.x * blockDim.x + threadIdx.x;
  if (i < n) deg[i] = 1.0f;  // self-loop
}

__global__ void k_count_deg(const int* __restrict__ dst, float* deg, int e) {
  int i = blockIdx.x * blockDim.x + threadIdx.x;
  if (i < e) atomicAdd(&deg[dst[i]], 1.0f);
}

__global__ void k_rsqrt(float* d, int n) {
  int i = blockIdx.x * blockDim.x + threadIdx.x;
  if (i < n) d[i] = rsqrtf(d[i]);  // deg >= 1 always
}

// ---------------- WMMA GEMM, K = 128 fixed, LDB compile-time ----------------
// C[M x nvalid] = A[M x 128] * B[128 x nvalid]  (B row-major, ld = LDB)
// epi: 0 = raw store, 1 = +bias then ELU, 2 = +bias
// Block: 256 threads = 8 waves; block covers 16 rows x (8*16) cols.
// Padded columns: B column index CLAMPED in-bounds (each output column depends
// only on its own B column; clamped lanes compute garbage that is never
// stored). All B offsets are compile-time constants off one 32-bit lane base
// -> pure immediate-offset global_load_b32, zero per-step address math.
template <int LDB>
__global__ void k_wmma_gemm_k128(const float* __restrict__ A,
                                 const float* __restrict__ B,
                                 const float* __restrict__ bias,
                                 float* __restrict__ C, int ldc,
                                 int M, int nvalid, int epi) {
  __shared__ float As[16 * DFEAT];  // 8 KB A tile shared by all 8 waves
  const int m0 = blockIdx.x * 16;

  // stage 16x128 A tile into LDS (float4 loads)
  for (int t = threadIdx.x; t < 16 * (DFEAT / 4); t += blockDim.x) {
    int r  = t >> 5;          // row 0..15
    int c4 = (t & 31) << 2;   // col 0..124 step 4
    int rr = m0 + r;
    if (rr >= M) rr = M - 1;  // clamp (avoid OOB read on ragged tail)
    *(float4*)(&As[r * DFEAT + c4]) =
        *(const float4*)(&A[(size_t)rr * DFEAT + c4]);
  }
  __syncthreads();

  const int lane  = threadIdx.x & 31;
  const int wave  = threadIdx.x >> 5;
  const int n0    = wave * 16;
  const int ntile = (nvalid + 15) & ~15;
  if (n0 >= ntile) return;  // wave-uniform exit; active waves keep EXEC all-1s

  const int mr    = lane & 15;          // M within tile (A) / N within tile (B)
  const int khalf = (lane >> 4) << 1;   // 0 or 2: half-wave K-pair select
  const int n     = n0 + mr;
  const int ncl   = n < nvalid ? n : (nvalid - 1);  // clamped, always legal

  // lane-dependent 32-bit base index; all per-step offsets are constants
  const int   bbase = khalf * LDB + ncl;
  const float* ap   = As + mr * DFEAT + khalf;

  v8f acc = {};
#pragma unroll
  for (int kk = 0; kk < DFEAT; kk += 4) {
    v2f a, b;
    a.x = ap[kk];
    a.y = ap[kk + 1];
    b.x = B[bbase + kk * LDB];          // immediate offset: kk*LDB*4 bytes
    b.y = B[bbase + (kk + 1) * LDB];    // immediate offset
    acc = __builtin_amdgcn_wmma_f32_16x16x4_f32(false, a, false, b,
                                                (short)0, acc, false, false);
  }

  if (n >= nvalid) return;  // padded column: nothing to store
  const int mbase = m0 + ((lane >> 4) << 3);  // C/D: VGPR g -> M = g (+8 hi half)
#pragma unroll
  for (int g = 0; g < 8; ++g) {
    int m = mbase + g;
    if (m >= M) continue;
    float v = acc[g];
    if (epi >= 1) v += bias[n];
    if (epi == 1) v = v > 0.0f ? v : expm1f(v);  // ELU(alpha=1)
    C[(size_t)m * ldc + n] = v;
  }
}

// ---------------- aggregation ----------------
// agg[i][:] = t[i][:] * dis[i]^2 + bias   (self-loop term + bias)
__global__ void k_agg_init(const float* __restrict__ t,
                           const float* __restrict__ dis,
                           const float* __restrict__ bias,
                           float* __restrict__ agg, int n) {
  long long tid = (long long)blockIdx.x * blockDim.x + threadIdx.x;
  int i = (int)(tid >> 5);
  if (i >= n) return;
  int c = (int)(tid & 31) << 2;
  float di = dis[i];
  float s  = di * di;
  float4 v  = *(const float4*)(t + (size_t)i * DFEAT + c);
  float4 bb = *(const float4*)(bias + c);
  float4 o;
  o.x = v.x * s + bb.x;
  o.y = v.y * s + bb.y;
  o.z = v.z * s + bb.z;
  o.w = v.w * s + bb.w;
  *(float4*)(agg + (size_t)i * DFEAT + c) = o;
}

// agg[dst[e]][:] += t[src[e]][:] * dis[src]*dis[dst]   (L2-resident atomics)
__global__ void k_scatter(const float* __restrict__ t,
                          const int* __restrict__ src,
                          const int* __restrict__ dst,
                          const float* __restrict__ dis,
                          float* agg, int e) {
  long long tid = (long long)blockIdx.x * blockDim.x + threadIdx.x;
  int ed = (int)(tid >> 5);
  if (ed >= e) return;
  int c = (int)(tid & 31) << 2;
  int s = src[ed];
  int d = dst[ed];
  float nrm = dis[s] * dis[d];
  float4 v = *(const float4*)(t + (size_t)s * DFEAT + c);
  float* o = agg + (size_t)d * DFEAT + c;
  atomicAdd(o + 0, v.x * nrm);
  atomicAdd(o + 1, v.y * nrm);
  atomicAdd(o + 2, v.z * nrm);
  atomicAdd(o + 3, v.w * nrm);
}

// ReLU in place; optionally also mirror to emb output
__global__ void k_relu(float* __restrict__ x, float* __restrict__ emb,
                       long long n) {
  long long i = (long long)blockIdx.x * blockDim.x + threadIdx.x;
  if (i >= n) return;
  float v = fmaxf(x[i], 0.0f);
  x[i] = v;
  if (emb) emb[i] = v;
}

// ---------------- softmax over 40 classes ----------------
__global__ void k_softmax40(const float* __restrict__ logits,
                            float* __restrict__ probs, int n) {
  int i = blockIdx.x * blockDim.x + threadIdx.x;
  if (i >= n) return;
  const float* l = logits + (size_t)i * 40;
  float mx = -3.402823466e+38f;
#pragma unroll
  for (int j = 0; j < 40; ++j) mx = fmaxf(mx, l[j]);
  float e[40];
  float sum = 0.0f;
#pragma unroll
  for (int j = 0; j < 40; ++j) {
    float t = expf(l[j] - mx);
    e[j] = t;
    sum += t;
  }
  float inv = 1.0f / sum;
  float* p = probs + (size_t)i * 40;
#pragma unroll
  for (int j = 0; j < 40; ++j) p[j] = e[j] * inv;
}

// ---------------- driver ----------------
extern "C" void kernel_launch(void* const* d_in, const int* in_sizes, int n_in,
                              void* d_out, int out_size, void* d_ws, size_t ws_size,
                              hipStream_t stream) {
  const float* x   = (const float*)d_in[0];
  const int*   ei  = (const int*)d_in[1];   // [2, E]: row 0 = src, row 1 = dst
  const float* W1  = (const float*)d_in[2];
  const float* b1  = (const float*)d_in[3];
  const float* W2  = (const float*)d_in[4];
  const float* b2  = (const float*)d_in[5];
  const float* W3  = (const float*)d_in[6];
  const float* b3  = (const float*)d_in[7];
  const float* Wm1 = (const float*)d_in[8];
  const float* bm1 = (const float*)d_in[9];
  const float* Wm2 = (const float*)d_in[10];
  const float* bm2 = (const float*)d_in[11];

  const int N = in_sizes[0] / DFEAT;
  const int E = in_sizes[1] / 2;
  const int* src = ei;
  const int* dst = ei + E;

  float* dis  = (float*)d_ws;                 // N
  float* bufA = dis + N;                      // N*128 (16B-aligned: N%4==0)
  float* bufB = bufA + (size_t)N * DFEAT;     // N*128

  float* logits = (float*)d_out;
  float* probs  = logits + (size_t)N * 40;
  float* emb    = probs  + (size_t)N * 40;

  const int T = 256;
  // --- degree -> dis = rsqrt(indegree + 1) ---
  k_init_deg<<<(N + T - 1) / T, T, 0, stream>>>(dis, N);
  k_count_deg<<<(E + T - 1) / T, T, 0, stream>>>(dst, dis, E);
  k_rsqrt<<<(N + T - 1) / T, T, 0, stream>>>(dis, N);

  const int gemmBlocks = (N + 15) / 16;
  const long long nodeThreads = (long long)N * 32;
  const long long edgeThreads = (long long)E * 32;
  const long long featElems   = (long long)N * DFEAT;
  const int nodeBlocks = (int)((nodeThreads + T - 1) / T);
  const int edgeBlocks = (int)((edgeThreads + T - 1) / T);
  const int featBlocks = (int)((featElems + T - 1) / T);

  const float* cur = x;
  const float* Ws[3] = {W1, W2, W3};
  const float* bs[3] = {b1, b2, b3};
  for (int l = 0; l < 3; ++l) {
    // t = cur @ W  (WMMA)
    k_wmma_gemm_k128<DFEAT><<<gemmBlocks, 256, 0, stream>>>(
        cur, Ws[l], nullptr, bufA, DFEAT, N, DFEAT, 0);
    // agg = t * dis^2 + b (self loop + bias)
    k_agg_init<<<nodeBlocks, T, 0, stream>>>(bufA, dis, bs[l], bufB, N);
    // agg[dst] += t[src] * dis[src]*dis[dst]
    k_scatter<<<edgeBlocks, T, 0, stream>>>(bufA, src, dst, dis, bufB, E);
    // ReLU (last layer also emits emb)
    k_relu<<<featBlocks, T, 0, stream>>>(bufB, (l == 2) ? emb : nullptr, featElems);
    cur = bufB;
  }

  // MLP head: m = ELU(emb @ Wm1 + bm1)
  k_wmma_gemm_k128<DFEAT><<<gemmBlocks, 256, 0, stream>>>(
      bufB, Wm1, bm1, bufA, DFEAT, N, DFEAT, 1);
  // logits = m @ Wm2 + bm2  (40 cols, padded tile with clamped loads)
  k_wmma_gemm_k128<40><<<gemmBlocks, 256, 0, stream>>>(
      bufA, Wm2, bm2, logits, 40, N, 40, 2);
  // probs = softmax(logits)
  k_softmax40<<<(N + T - 1) / T, T, 0, stream>>>(logits, probs, N);
}